// GraphAttentionNetwork_3221225472508
// MI455X (gfx1250) — compile-verified
//
#include <hip/hip_runtime.h>
#include <hip/hip_bf16.h>

// ---------------------------------------------------------------------------
// GAT on MI455X (gfx1250): FP32 WMMA (v_wmma_f32_16x16x4_f32) for all dense
// GEMMs (A-fragment reused across all N tiles), global_atomic_add_f32 for the
// per-target segment sums over edges.
// ---------------------------------------------------------------------------

#define FEAT    128
#define UNITS   8
#define HEADS   8
#define HIDDEN  64        // UNITS*HEADS
#define OUT_DIM 16
#define NCOMB   80        // HIDDEN + 2*HEADS (h | a_t | a_s)

typedef __attribute__((ext_vector_type(2))) float v2f;
typedef __attribute__((ext_vector_type(8))) float v8f;

// ---------------------------------------------------------------------------
// FP32 WMMA GEMM:  C[M,N] = act(A[M,K] @ B[K,N] + bias),  N == NT*16, K%4==0.
// 256-thread blocks = 8 waves; each wave owns one 16-row strip and computes
// all NT column tiles, so every A fragment is loaded once and fed to NT
// back-to-back v_wmma ops.
// A-frag: lane l -> row l%16, K = (l/16)*2 + {0,1}   (contiguous float2)
// B-frag: lane l -> col l%16, K = (l/16)*2 + {0,1}
// C-frag: vgpr i -> row = strip*16 + i + (l/16)*8, col = tile*16 + l%16
// ---------------------------------------------------------------------------
template <int NT>
__global__ __launch_bounds__(256) void gemm_f32_wmma(
    const float* __restrict__ A, const float* __restrict__ B,
    const float* __restrict__ bias, float* __restrict__ C,
    int M, int K, int doRelu)
{
    constexpr int N = NT * 16;
    const int wave = threadIdx.x >> 5;
    const int lane = threadIdx.x & 31;
    const int half = lane >> 4;     // 0: K={0,1}, 1: K={2,3} within each step
    const int l16  = lane & 15;

    const int strip = blockIdx.x * 8 + wave;     // 16-row strip index
    if (strip * 16 >= M) return;                 // whole wave exits together

    const int arow = strip * 16 + l16;
    const float* Ap = A + (size_t)arow * K + (half << 1);
    const float* Bp = B + (size_t)(half << 1) * N + l16;

    v8f acc[NT];
#pragma unroll
    for (int t = 0; t < NT; ++t) acc[t] = (v8f){};

    for (int k = 0; k < K; k += 4) {
        v2f a;
        a.x = Ap[0];       // A[arow][k + half*2 + 0]
        a.y = Ap[1];       // A[arow][k + half*2 + 1]
#pragma unroll
        for (int t = 0; t < NT; ++t) {
            v2f b;
            b.x = Bp[t * 16];        // B[k + half*2 + 0][t*16 + l16]
            b.y = Bp[t * 16 + N];    // B[k + half*2 + 1][t*16 + l16]
            acc[t] = __builtin_amdgcn_wmma_f32_16x16x4_f32(
                /*neg_a=*/false, a, /*neg_b=*/false, b,
                /*c_mod=*/(short)0, acc[t], /*reuse_a=*/false, /*reuse_b=*/false);
        }
        Ap += 4;
        Bp += (size_t)4 * N;
    }

#pragma unroll
    for (int t = 0; t < NT; ++t) {
        const int col = t * 16 + l16;
        const float bv = bias ? bias[col] : 0.0f;
#pragma unroll
        for (int i = 0; i < 8; ++i) {
            float v = acc[t][i] + bv;
            if (doRelu) v = fmaxf(v, 0.0f);
            const int row = strip * 16 + i + (half << 3);
            C[(size_t)row * N + col] = v;
        }
    }
}

// ---------------------------------------------------------------------------
// Build combined weight Wcomb[64,80] for one layer:
//   cols 0..63  : K_flat[d, h*8+u]            = gat_kernel[l,h,d,u]
//   cols 64..71 : KA1[d,h] = sum_u K[l,h,d,u] * gat_attn[l,h,u]
//   cols 72..79 : KA2[d,h] = sum_u K[l,h,d,u] * gat_attn[l,h,8+u]
// ---------------------------------------------------------------------------
__global__ void build_wcomb(const float* __restrict__ gk,
                            const float* __restrict__ ga,
                            float* __restrict__ w, int layer)
{
    int idx = blockIdx.x * blockDim.x + threadIdx.x;
    if (idx >= HIDDEN * NCOMB) return;
    const int d = idx / NCOMB;
    const int c = idx % NCOMB;
    float v;
    if (c < HIDDEN) {
        const int h = c >> 3, u = c & 7;
        v = gk[(((size_t)layer * HEADS + h) * HIDDEN + d) * UNITS + u];
    } else {
        const int second = (c >= HIDDEN + HEADS);
        const int h = c - HIDDEN - (second ? HEADS : 0);
        const float* kp = gk + (((size_t)layer * HEADS + h) * HIDDEN + d) * UNITS;
        const float* ap = ga + ((size_t)layer * HEADS + h) * (2 * UNITS) + (second ? UNITS : 0);
        float acc = 0.0f;
#pragma unroll
        for (int u = 0; u < UNITS; ++u) acc += kp[u] * ap[u];
        v = acc;
    }
    w[(size_t)d * NCOMB + c] = v;
}

__global__ void zero_f32v4(float4* __restrict__ p, int n4)
{
    int idx = blockIdx.x * blockDim.x + threadIdx.x;
    if (idx < n4) p[idx] = make_float4(0.f, 0.f, 0.f, 0.f);
}

__device__ __forceinline__ float gat_score(float raw)
{
    raw = (raw >= 0.0f) ? raw : 0.2f * raw;       // LeakyReLU(0.2)
    raw = fminf(fmaxf(raw, -2.0f), 2.0f);         // clip
    return __expf(raw);
}

// ---------------------------------------------------------------------------
// Pass 1, one thread per EDGE: scores for all 8 heads + atomic ssum[tgt,h].
// a_t / a_s slices are 16B-aligned (row stride 320B, offsets 256B / 288B).
// ---------------------------------------------------------------------------
__global__ void edge_scores(const long long* __restrict__ edges,
                            const float* __restrict__ hcomb,
                            float* __restrict__ scores,
                            float* __restrict__ ssum, int nE)
{
    int e = blockIdx.x * blockDim.x + threadIdx.x;
    if (e >= nE) return;
    const int t = (int)edges[2 * (size_t)e + 0];
    const int s = (int)edges[2 * (size_t)e + 1];

    const float4* at = (const float4*)(hcomb + (size_t)t * NCOMB + HIDDEN);
    const float4* as = (const float4*)(hcomb + (size_t)s * NCOMB + HIDDEN + HEADS);
    const float4 a0 = at[0], a1 = at[1];
    const float4 b0 = as[0], b1 = as[1];

    float4 s0, s1;
    s0.x = gat_score(a0.x + b0.x);
    s0.y = gat_score(a0.y + b0.y);
    s0.z = gat_score(a0.z + b0.z);
    s0.w = gat_score(a0.w + b0.w);
    s1.x = gat_score(a1.x + b1.x);
    s1.y = gat_score(a1.y + b1.y);
    s1.z = gat_score(a1.z + b1.z);
    s1.w = gat_score(a1.w + b1.w);

    float4* sp = (float4*)(scores + (size_t)e * HEADS);
    sp[0] = s0;
    sp[1] = s1;

    float* sm = ssum + (size_t)t * HEADS;
    atomicAdd(&sm[0], s0.x); atomicAdd(&sm[1], s0.y);
    atomicAdd(&sm[2], s0.z); atomicAdd(&sm[3], s0.w);
    atomicAdd(&sm[4], s1.x); atomicAdd(&sm[5], s1.y);
    atomicAdd(&sm[6], s1.z); atomicAdd(&sm[7], s1.w);
}

// ---------------------------------------------------------------------------
// Pass 2, one thread per EDGE: alpha[h] = score/ssum[tgt,h];
// agg[tgt, h*8+u] += alpha[h] * h[src, h*8+u]   (64 float atomics per edge)
// ---------------------------------------------------------------------------
__global__ void edge_aggregate(const long long* __restrict__ edges,
                               const float* __restrict__ hcomb,
                               const float* __restrict__ scores,
                               const float* __restrict__ ssum,
                               float* __restrict__ agg, int nE)
{
    int e = blockIdx.x * blockDim.x + threadIdx.x;
    if (e >= nE) return;
    const int t = (int)edges[2 * (size_t)e + 0];
    const int s = (int)edges[2 * (size_t)e + 1];

    const float4* scv = (const float4*)(scores + (size_t)e * HEADS);
    const float4* smv = (const float4*)(ssum + (size_t)t * HEADS);
    const float4 sc0 = scv[0], sc1 = scv[1];
    const float4 sm0 = smv[0], sm1 = smv[1];
    float alpha[HEADS];
    alpha[0] = sc0.x / sm0.x; alpha[1] = sc0.y / sm0.y;
    alpha[2] = sc0.z / sm0.z; alpha[3] = sc0.w / sm0.w;
    alpha[4] = sc1.x / sm1.x; alpha[5] = sc1.y / sm1.y;
    alpha[6] = sc1.z / sm1.z; alpha[7] = sc1.w / sm1.w;

    const float* hv = hcomb + (size_t)s * NCOMB;   // h values, cols 0..63
    float*       ov = agg   + (size_t)t * HIDDEN;
#pragma unroll
    for (int h = 0; h < HEADS; ++h) {
        const float4* h4 = (const float4*)(hv + h * UNITS);
        const float4 v0 = h4[0], v1 = h4[1];
        const float a = alpha[h];
        float* o = ov + h * UNITS;
        atomicAdd(&o[0], a * v0.x); atomicAdd(&o[1], a * v0.y);
        atomicAdd(&o[2], a * v0.z); atomicAdd(&o[3], a * v0.w);
        atomicAdd(&o[4], a * v1.x); atomicAdd(&o[5], a * v1.y);
        atomicAdd(&o[6], a * v1.z); atomicAdd(&o[7], a * v1.w);
    }
}

// x = relu(agg) + x   (residual), float4-vectorized
__global__ void finalize_layer(const float4* __restrict__ agg,
                               float4* __restrict__ x, int n4)
{
    int idx = blockIdx.x * blockDim.x + threadIdx.x;
    if (idx >= n4) return;
    const float4 a = agg[idx];
    float4 v = x[idx];
    v.x += fmaxf(a.x, 0.0f);
    v.y += fmaxf(a.y, 0.0f);
    v.z += fmaxf(a.z, 0.0f);
    v.w += fmaxf(a.w, 0.0f);
    x[idx] = v;
}

// ---------------------------------------------------------------------------
extern "C" void kernel_launch(void* const* d_in, const int* in_sizes, int n_in,
                              void* d_out, int out_size, void* d_ws, size_t ws_size,
                              hipStream_t stream)
{
    const float*     node_states = (const float*)d_in[0];     // [50000,128]
    const long long* edges       = (const long long*)d_in[1]; // [800000,2] int64
    const float*     W_pre       = (const float*)d_in[2];     // [128,64]
    const float*     b_pre       = (const float*)d_in[3];     // [64]
    const float*     gat_kernel  = (const float*)d_in[4];     // [2,8,64,8]
    const float*     gat_attn    = (const float*)d_in[5];     // [2,8,16]
    const float*     W_out       = (const float*)d_in[6];     // [64,16]
    const float*     b_out       = (const float*)d_in[7];     // [16]
    float*           out         = (float*)d_out;             // [50000,16]

    const int M  = in_sizes[0] / FEAT;       // 50000 (divisible by 16)
    const int nE = in_sizes[1] / 2;          // 800000

    // Workspace layout (floats)
    float* ws     = (float*)d_ws;
    float* xbuf   = ws;                                   // [M,64]
    float* hcomb  = xbuf  + (size_t)M * HIDDEN;           // [M,80]
    float* scores = hcomb + (size_t)M * NCOMB;            // [nE,8]
    float* ssum   = scores + (size_t)nE * HEADS;          // [M,8]
    float* agg    = ssum  + (size_t)M * HEADS;            // [M,64]
    float* wcomb  = agg   + (size_t)M * HIDDEN;           // [64,80]

    const int strips = M / 16;               // 3125 row strips
    const int gemmBlocks = (strips + 7) / 8; // 8 waves (strips) per block

    // 1) Pre-projection: xbuf = relu(node_states @ W_pre + b_pre)   N=64
    gemm_f32_wmma<4><<<gemmBlocks, 256, 0, stream>>>(
        node_states, W_pre, b_pre, xbuf, M, FEAT, 1);

    for (int l = 0; l < 2; ++l) {
        // 2a) Combined weight [64,80] = [K_flat | K·A1 | K·A2]
        build_wcomb<<<(HIDDEN * NCOMB + 255) / 256, 256, 0, stream>>>(
            gat_kernel, gat_attn, wcomb, l);

        // 2b) hcomb = xbuf @ wcomb  (h in cols 0..63, a_t 64..71, a_s 72..79)
        gemm_f32_wmma<5><<<gemmBlocks, 256, 0, stream>>>(
            xbuf, wcomb, nullptr, hcomb, M, HIDDEN, 0);

        // 2c) Zero segment-sum and aggregation buffers
        zero_f32v4<<<((M * HEADS / 4) + 255) / 256, 256, 0, stream>>>(
            (float4*)ssum, M * HEADS / 4);
        zero_f32v4<<<((M * HIDDEN / 4) + 255) / 256, 256, 0, stream>>>(
            (float4*)agg, M * HIDDEN / 4);

        // 2d) Edge scores + per-target softmax denominators
        edge_scores<<<(nE + 255) / 256, 256, 0, stream>>>(
            edges, hcomb, scores, ssum, nE);

        // 2e) Alpha-weighted aggregation into agg
        edge_aggregate<<<(nE + 255) / 256, 256, 0, stream>>>(
            edges, hcomb, scores, ssum, agg, nE);

        // 2f) Residual: xbuf = relu(agg) + xbuf
        finalize_layer<<<((M * HIDDEN / 4) + 255) / 256, 256, 0, stream>>>(
            (const float4*)agg, (float4*)xbuf, M * HIDDEN / 4);
    }

    // 3) Output projection: out = xbuf @ W_out + b_out   N=16
    gemm_f32_wmma<1><<<gemmBlocks, 256, 0, stream>>>(
        xbuf, W_out, b_out, out, M, HIDDEN, 0);
}